// MambaLayer_39968965657052
// MI455X (gfx1250) — compile-verified
//
#include <hip/hip_runtime.h>

#define CH   1536
#define SL   2048
#define NST  16          // state size n
#define TILE 64          // timesteps per LDS tile
#define NTILES (SL / TILE)
#define ROWS 16          // channels per block
#define RSTRIDE 68       // LDS row stride in floats (TDM pads 4 dwords / 64 dwords)

typedef __attribute__((ext_vector_type(4))) unsigned int v4u;
typedef __attribute__((ext_vector_type(8))) int v8i;
typedef __attribute__((ext_vector_type(4))) int v4i;

__device__ __forceinline__ unsigned lds_byte_off(const void* p) {
  // flat LDS aperture address: low 32 bits are the LDS byte address
  return (unsigned)(unsigned long long)p;
}

// 2D tile load via Tensor Data Mover: `rows` rows of TILE floats, row stride
// SL elements, with LDS padding of 4 dwords after every 64 dwords so the LDS
// row stride becomes 68 floats (bank-conflict-free column access: bank 4n+t).
__device__ __forceinline__ void tdm_load_tile(const float* gsrc, unsigned ldsOff,
                                              unsigned rows, unsigned tensorRows) {
  unsigned long long ga = (unsigned long long)gsrc;
  v4u g0;
  g0.x = 1u;                                        // count=1, is_restore=0
  g0.y = ldsOff;                                    // lds_addr (bytes)
  g0.z = (unsigned)ga;                              // global_addr[31:0]
  g0.w = ((unsigned)(ga >> 32) & 0x01FFFFFFu)       // global_addr[56:32]
         | (2u << 30);                              // type = 2 (image)
  v8i g1;
  g1[0] = (int)((2u << 16)                          // data_size = 4B
              | (1u << 20)                          // pad_enable
              | (5u << 22)                          // pad_interval: 64 dwords
              | (3u << 25));                        // pad_amount: 4 dwords
  g1[1] = (int)(((unsigned)SL & 0xFFFFu) << 16);    // tensor_dim0[15:0]
  g1[2] = (int)(((unsigned)SL >> 16)                // tensor_dim0[31:16]
              | ((tensorRows & 0xFFFFu) << 16));    // tensor_dim1[15:0]
  g1[3] = (int)(((tensorRows >> 16) & 0xFFFFu)      // tensor_dim1[31:16]
              | ((unsigned)TILE << 16));            // tile_dim0 = 64
  g1[4] = (int)(rows & 0xFFFFu);                    // tile_dim1; tile_dim2=0
  g1[5] = (int)SL;                                  // tensor_dim0_stride[31:0]
  g1[6] = 0;                                        // stride hi / dim1_stride lo
  g1[7] = 0;
  v4i g2 = {0, 0, 0, 0};                            // 2D: groups 2/3 unused
  v4i g3 = {0, 0, 0, 0};
  v8i g4 = {0, 0, 0, 0, 0, 0, 0, 0};
  __builtin_amdgcn_tensor_load_to_lds(g0, g1, g2, g3, g4, 0);
}

// 16-lane XOR-butterfly sum on the VALU (v_permlane16_b32), keeping the LDS
// pipe free for tile traffic. Reduces both 16-lane halves of the wave at once.
__device__ __forceinline__ float reduce16(float p) {
#if __has_builtin(__builtin_amdgcn_permlane16)
  unsigned v = __float_as_uint(p);
  // sel patterns: nibble i holds (i ^ mask) for lanes 0..7 (s0) and 8..15 (s1)
  unsigned t;
  t = __builtin_amdgcn_permlane16(v, v, 0x67452301u, 0xEFCDAB89u, false, false); // i^1
  v = __float_as_uint(__uint_as_float(v) + __uint_as_float(t));
  t = __builtin_amdgcn_permlane16(v, v, 0x54761032u, 0xDCFE98BAu, false, false); // i^2
  v = __float_as_uint(__uint_as_float(v) + __uint_as_float(t));
  t = __builtin_amdgcn_permlane16(v, v, 0x32107654u, 0xBA98FEDCu, false, false); // i^4
  v = __float_as_uint(__uint_as_float(v) + __uint_as_float(t));
  t = __builtin_amdgcn_permlane16(v, v, 0xFEDCBA98u, 0x76543210u, false, false); // i^8
  return __uint_as_float(v) + __uint_as_float(t);
#else
  p += __shfl_xor(p, 1, 32);
  p += __shfl_xor(p, 2, 32);
  p += __shfl_xor(p, 4, 32);
  p += __shfl_xor(p, 8, 32);
  return p;
#endif
}

__global__ __launch_bounds__(256)
void MambaLayer_39968965657052_kernel(const float* __restrict__ delta,
                                      const float* __restrict__ A,
                                      const float* __restrict__ B,
                                      const float* __restrict__ u,
                                      const float* __restrict__ C,
                                      float* __restrict__ out) {
  __shared__ float s_d[2][ROWS * RSTRIDE + 16];
  __shared__ float s_u[2][ROWS * RSTRIDE + 16];
  __shared__ float s_B[2][NST * RSTRIDE + 16];
  __shared__ float s_C[2][NST * RSTRIDE + 16];
  __shared__ float s_out[ROWS * TILE];

  const int b    = blockIdx.y;
  const int c0   = blockIdx.x * ROWS;
  const int tid  = threadIdx.x;
  const int wave = tid >> 5;
  const int lane = tid & 31;
  const int cRow = (wave << 1) | (lane >> 4);   // 0..15: channel within block
  const int n    = lane & 15;                   // state index

  const float a = A[(c0 + cRow) * NST + n];

  const float* dBase = delta + (size_t)(b * CH + c0) * SL;
  const float* uBase = u     + (size_t)(b * CH + c0) * SL;
  const float* BBase = B + (size_t)b * NST * SL;
  const float* CBase = C + (size_t)b * NST * SL;
  float*       oBase = out + (size_t)(b * CH + c0) * SL;

  if (wave == 0) {
    tdm_load_tile(dBase, lds_byte_off(&s_d[0][0]), ROWS, CH);
    tdm_load_tile(uBase, lds_byte_off(&s_u[0][0]), ROWS, CH);
    tdm_load_tile(BBase, lds_byte_off(&s_B[0][0]), NST, NST);
    tdm_load_tile(CBase, lds_byte_off(&s_C[0][0]), NST, NST);
  }

  float h = 0.0f;   // recurrence state for this (b, c, n), carried across tiles

  for (int it = 0; it < NTILES; ++it) {
    const int buf = it & 1;
    if (wave == 0) {
      if (it + 1 < NTILES) {
        const int nb = (it + 1) & 1;
        const float* dt = dBase + (it + 1) * TILE;
        const float* ut = uBase + (it + 1) * TILE;
        const float* Bt = BBase + (it + 1) * TILE;
        const float* Ct = CBase + (it + 1) * TILE;
        tdm_load_tile(dt, lds_byte_off(&s_d[nb][0]), ROWS, CH);
        tdm_load_tile(ut, lds_byte_off(&s_u[nb][0]), ROWS, CH);
        tdm_load_tile(Bt, lds_byte_off(&s_B[nb][0]), NST, NST);
        tdm_load_tile(Ct, lds_byte_off(&s_C[nb][0]), NST, NST);
        __builtin_amdgcn_s_wait_tensorcnt(4);   // tile `it` retired (in-order)
      } else {
        __builtin_amdgcn_s_wait_tensorcnt(0);
      }
    }
    __syncthreads();

    const float* dR = &s_d[buf][cRow * RSTRIDE];
    const float* uR = &s_u[buf][cRow * RSTRIDE];
    const float* bR = &s_B[buf][n * RSTRIDE];
    const float* cR = &s_C[buf][n * RSTRIDE];

    #pragma unroll
    for (int tt = 0; tt < TILE; tt += 4) {
      const float4 d4 = *(const float4*)(dR + tt);
      const float4 u4 = *(const float4*)(uR + tt);
      const float4 b4 = *(const float4*)(bR + tt);
      const float4 c4 = *(const float4*)(cR + tt);
      const float dv[4] = {d4.x, d4.y, d4.z, d4.w};
      const float uv[4] = {u4.x, u4.y, u4.z, u4.w};
      const float bv[4] = {b4.x, b4.y, b4.z, b4.w};
      const float cv[4] = {c4.x, c4.y, c4.z, c4.w};
      float pv[4];
      #pragma unroll
      for (int j = 0; j < 4; ++j) {
        const float dA  = __expf(dv[j] * a);          // exp(delta * A[c,n])
        const float dbu = dv[j] * uv[j] * bv[j];      // delta * u * B
        h = fmaf(dA, h, dbu);                         // h = dA*h + dBu
        pv[j] = h * cv[j];
      }
      // four independent VALU butterflies; off the serial h critical path
      #pragma unroll
      for (int j = 0; j < 4; ++j) {
        const float p = reduce16(pv[j]);
        if (n == 0) s_out[cRow * TILE + tt + j] = p;
      }
    }
    __syncthreads();

    // coalesced float4 store of the 16x64 output tile
    {
      const int r  = tid >> 4;          // 0..15 channel row
      const int cc = (tid & 15) << 2;   // 0,4,...,60
      const float4 v = *(const float4*)&s_out[r * TILE + cc];
      *(float4*)(oBase + (size_t)r * SL + it * TILE + cc) = v;
    }
    // s_out / input-buffer reuse is fenced by the __syncthreads() at the top
    // of the next iteration (TDM issue for tile it+2 targets the buffer whose
    // readers all passed the post-compute barrier above).
  }
}

extern "C" void kernel_launch(void* const* d_in, const int* in_sizes, int n_in,
                              void* d_out, int out_size, void* d_ws, size_t ws_size,
                              hipStream_t stream) {
  const float* delta = (const float*)d_in[0];
  const float* A     = (const float*)d_in[1];
  const float* B     = (const float*)d_in[2];
  const float* u     = (const float*)d_in[3];
  const float* C     = (const float*)d_in[4];
  float* out = (float*)d_out;

  const int batch = in_sizes[0] / (CH * SL);   // = 2
  dim3 grid(CH / ROWS, batch);
  MambaLayer_39968965657052_kernel<<<grid, 256, 0, stream>>>(delta, A, B, u, C, out);
}